// IsoMaxPlusLossFirstPart_40578851012977
// MI455X (gfx1250) — compile-verified
//
#include <hip/hip_runtime.h>
#include <math.h>

typedef float v2f __attribute__((ext_vector_type(2)));
typedef float v8f __attribute__((ext_vector_type(8)));

#define B_ 2
#define C_ 128
#define H_ 128
#define W_ 256
#define K_ 20
#define KP 32
#define CS 132   // tileT row stride (C floats padded) -> conflict-free b64 A loads
#define PS 132   // prototype row stride
#define OS 261   // output transpose-tile row stride (conflict-free for D layout)

// LDS float offsets
#define OFF_TILE 0
#define OFF_PN   (W_ * CS)            // 33792
#define OFF_CINV (OFF_PN + KP * PS)   // 38016
#define OFF_QV   (OFF_CINV + C_)      // 38144
#define OFF_PINV (OFF_QV + W_)        // 38400
#define OFF_SCR  (OFF_PINV + 32)      // 38432
#define LDS_FLOATS (OFF_SCR + 256)    // 38688
#define LDS_BYTES  (LDS_FLOATS * 4)   // 154752 bytes (< 320KB WGP LDS)

extern "C" __global__ __launch_bounds__(256)
void isomax_logits_kernel(const float* __restrict__ feat,
                          const float* __restrict__ proto,
                          const float* __restrict__ dscale,
                          float* __restrict__ out)
{
    extern __shared__ float lds[];
    const int t  = threadIdx.x;
    const int bh = blockIdx.x;
    const int b  = bh / H_;
    const int h  = bh % H_;

    const float absds = fabsf(dscale[0]);

    // ---- P0: stream feature slice [C][W] -> LDS transposed tileT[w][c]; zero pn pad
    {
        const float* fslice = feat + ((size_t)(b * C_) * H_ + h) * (size_t)W_;
        #pragma unroll 4
        for (int e = t; e < C_ * W_; e += 256) {
            int c = e >> 8;          // W_ == 256
            int w = e & 255;
            lds[OFF_TILE + w * CS + c] = fslice[(size_t)c * (H_ * W_) + w];
        }
        for (int e = t; e < KP * PS; e += 256)
            lds[OFF_PN + e] = 0.0f;
    }
    __syncthreads();

    // ---- P1: load raw prototypes; partial row-norms over W per channel c
    {
        for (int e = t; e < K_ * C_; e += 256) {
            int k = e >> 7;
            int c = e & 127;
            lds[OFF_PN + k * PS + c] = proto[e];
        }
        int c  = t & 127;
        int hw = t >> 7;             // 0 or 1: which half of W
        float s = 0.f;
        #pragma unroll 8
        for (int w = hw * 128; w < hw * 128 + 128; ++w) {
            float v = lds[OFF_TILE + w * CS + c];
            s += v * v;
        }
        lds[OFF_SCR + t] = s;
    }
    __syncthreads();

    // ---- P2: finalize 1/max(norm,eps) for channels and prototypes
    if (t < C_) {
        float n = sqrtf(lds[OFF_SCR + t] + lds[OFF_SCR + t + 128]);
        lds[OFF_CINV + t] = 1.0f / fmaxf(n, 1e-12f);
    } else if (t < 128 + K_) {
        int k = t - 128;
        float s = 0.f;
        #pragma unroll 8
        for (int c = 0; c < C_; ++c) {
            float v = lds[OFF_PN + k * PS + c];
            s += v * v;
        }
        lds[OFF_PINV + k] = 1.0f / fmaxf(sqrtf(s), 1e-12f);
    }
    __syncthreads();

    // ---- P3: scale tile columns by cinv[c]; scale prototype rows by pinv[k]
    {
        #pragma unroll 4
        for (int e = t; e < W_ * C_; e += 256) {
            int w = e >> 7;
            int c = e & 127;
            lds[OFF_TILE + w * CS + c] *= lds[OFF_CINV + c];
        }
        for (int e = t; e < K_ * C_; e += 256) {
            int k = e >> 7;
            int c = e & 127;
            lds[OFF_PN + k * PS + c] *= lds[OFF_PINV + k];
        }
    }
    __syncthreads();

    // ---- P4: q[w] = ||fn(:,w)||^2
    {
        float s = 0.f;
        #pragma unroll 8
        for (int c = 0; c < C_; ++c) {
            float v = lds[OFF_TILE + t * CS + c];
            s += v * v;
        }
        lds[OFF_QV + t] = s;
    }
    __syncthreads();

    // ---- P5: WMMA f32 16x16x4 GEMM  dot[w,k] = fn[w,:] . pn[k,:], fused distance math
    const int wave = t >> 5;
    const int lane = t & 31;
    const int l    = lane & 15;
    const int hf   = lane >> 4;

    float vals[4][8];
    int   tw0[4], tk0[4];
    #pragma unroll
    for (int i = 0; i < 4; ++i) {
        int idx = wave * 4 + i;                 // 32 tiles over 8 waves
        int w0  = (idx >> 1) * 16;              // 16 M-tiles of W
        int k0  = (idx & 1) * 16;               // 2 N-tiles of K (padded to 32)
        tw0[i] = w0; tk0[i] = k0;

        v8f acc = {0.f, 0.f, 0.f, 0.f, 0.f, 0.f, 0.f, 0.f};
        // ISA 16x4 f32 A layout: lanes 0-15 -> K = c0,c0+1 ; lanes 16-31 -> K = c0+2,c0+3
        const float* arow = &lds[OFF_TILE + (w0 + l) * CS + 2 * hf];
        const float* brow = &lds[OFF_PN   + (k0 + l) * PS + 2 * hf];
        #pragma unroll
        for (int c0 = 0; c0 < C_; c0 += 4) {
            v2f a  = *(const v2f*)(arow + c0);
            v2f bb = *(const v2f*)(brow + c0);
            acc = __builtin_amdgcn_wmma_f32_16x16x4_f32(false, a, false, bb,
                                                        (short)0, acc, false, false);
        }
        // D layout: lane l -> N = k0+l ; VGPR r -> M = w0 + 8*hf + r
        #pragma unroll
        for (int r = 0; r < 8; ++r) {
            int w   = w0 + 8 * hf + r;
            float q = lds[OFF_QV + w];
            float d2 = fmaxf(q + 1.0f - 2.0f * acc[r], 0.0f);
            vals[i][r] = -absds * sqrtf(d2);
        }
    }
    __syncthreads();   // all tileT reads complete -> safe to overlay

    // ---- transpose logits through LDS (overlays dead feature tile)
    #pragma unroll
    for (int i = 0; i < 4; ++i) {
        int krow  = tk0[i] + l;
        int wbase = tw0[i] + 8 * hf;
        #pragma unroll
        for (int r = 0; r < 8; ++r)
            lds[OFF_TILE + krow * OS + wbase + r] = vals[i][r];
    }
    __syncthreads();

    // ---- P6: coalesced global store: out[b][k][h][w]
    float* orow = out + (((size_t)b * K_) * H_ + h) * (size_t)W_;
    #pragma unroll
    for (int k = 0; k < K_; ++k)
        orow[(size_t)k * (H_ * W_) + t] = lds[OFF_TILE + k * OS + t];
}

extern "C" void kernel_launch(void* const* d_in, const int* in_sizes, int n_in,
                              void* d_out, int out_size, void* d_ws, size_t ws_size,
                              hipStream_t stream) {
    (void)in_sizes; (void)n_in; (void)out_size; (void)d_ws; (void)ws_size;
    const float* feat   = (const float*)d_in[0];
    const float* proto  = (const float*)d_in[1];
    const float* dscale = (const float*)d_in[2];
    float*       out    = (float*)d_out;

    // dynamic LDS > 64KB needs the attribute bump (host-side, capture-safe, idempotent)
    hipFuncSetAttribute((const void*)isomax_logits_kernel,
                        hipFuncAttributeMaxDynamicSharedMemorySize, LDS_BYTES);

    isomax_logits_kernel<<<dim3(B_ * H_), dim3(256), LDS_BYTES, stream>>>(
        feat, proto, dscale, out);
}